// GCN_68092411510979
// MI455X (gfx1250) — compile-verified
//
#include <hip/hip_runtime.h>

#define NN 100000
#define NE 1200000
#define D  64

typedef float v2f __attribute__((ext_vector_type(2)));
typedef float v8f __attribute__((ext_vector_type(8)));

// ---------------- utility kernels ----------------
__global__ void gcn_zero_kernel(float4* __restrict__ p, int n4) {
  int i = blockIdx.x * blockDim.x + threadIdx.x;
  if (i < n4) p[i] = make_float4(0.f, 0.f, 0.f, 0.f);
}

// deg layout: [0]=out(rel0,src0) [1]=in(rel0,dst0) [2]=out(rel1,src1) [3]=in(rel1,dst1)
__global__ void gcn_degree_kernel(const int* __restrict__ s0, const int* __restrict__ d0,
                                  const int* __restrict__ s1, const int* __restrict__ d1,
                                  float* __restrict__ deg) {
  int e = blockIdx.x * blockDim.x + threadIdx.x;
  if (e >= NE) return;
  atomicAdd(&deg[0 * NN + s0[e]], 1.f);
  atomicAdd(&deg[1 * NN + d0[e]], 1.f);
  atomicAdd(&deg[2 * NN + s1[e]], 1.f);
  atomicAdd(&deg[3 * NN + d1[e]], 1.f);
}

__global__ void gcn_rsqrt_kernel(float* __restrict__ deg, int n) {
  int i = blockIdx.x * blockDim.x + threadIdx.x;
  if (i < n) deg[i] = rsqrtf(fmaxf(deg[i], 1.f));   // zero-degree clamp as in DGL
}

// ---------------- H = (X @ W) * rdeg_out[:,None], via V_WMMA_F32_16X16X4_F32 ----------------
// One wave computes a 16-row x 64-col tile. W staged in LDS (16 KB).
// A frag (16x4 f32): lane<16 holds row=lane, K = k0+{0,1}; lane>=16 holds row=lane-16, K = k0+{2,3}.
// B frag (4x16 f32): vgpr0 = row k0+2*half, vgpr1 = row k0+2*half+1, col = lane%16.
// D (16x16 f32): vgpr v -> row v (lanes 0-15) / row v+8 (lanes 16-31), col = lane%16.
// All 16 A-fragment loads are issued before the WMMA stream so waits are partial.
__global__ void __launch_bounds__(256)
gcn_gemm_scale_kernel(const float* __restrict__ X, const float* __restrict__ W,
                      const float* __restrict__ rdeg, float* __restrict__ H) {
  __shared__ float Wl[D * D];
  for (int i = threadIdx.x; i < D * D; i += blockDim.x) Wl[i] = W[i];
  __syncthreads();

  const int lane = threadIdx.x & 31;
  const int half = lane >> 4;
  const int l16  = lane & 15;
  const int wave = (blockIdx.x * blockDim.x + threadIdx.x) >> 5;
  const int nwav = (gridDim.x * blockDim.x) >> 5;
  const int ntiles = NN / 16;  // 6250 exactly, no tail

  for (int t = wave; t < ntiles; t += nwav) {
    const int row0 = t * 16;
    const float* xrow = X + (size_t)(row0 + l16) * D + 2 * half;

    // Issue all A loads for the tile up front (16 x global_load_b64 in flight).
    v2f a[16];
#pragma unroll
    for (int k = 0; k < 16; ++k)
      a[k] = *(const v2f*)(xrow + 4 * k);

    v8f acc[4];
#pragma unroll
    for (int j = 0; j < 4; ++j) acc[j] = (v8f){0.f,0.f,0.f,0.f,0.f,0.f,0.f,0.f};

#pragma unroll
    for (int k = 0; k < 16; ++k) {
      const int kk = 4 * k + 2 * half;
#pragma unroll
      for (int j = 0; j < 4; ++j) {
        v2f b;
        b.x = Wl[kk * D + 16 * j + l16];
        b.y = Wl[(kk + 1) * D + 16 * j + l16];
        acc[j] = __builtin_amdgcn_wmma_f32_16x16x4_f32(
            /*neg_a=*/false, a[k], /*neg_b=*/false, b,
            /*c_mod=*/(short)0, acc[j], /*reuse_a=*/false, /*reuse_b=*/false);
      }
    }

#pragma unroll
    for (int v = 0; v < 8; ++v) {
      const int row = row0 + v + 8 * half;
      const float s = rdeg[row];
#pragma unroll
      for (int j = 0; j < 4; ++j)
        H[(size_t)row * D + 16 * j + l16] = acc[j][v] * s;
    }
  }
}

// ---------------- acc[dst] += H[src] * rdeg_in[dst]  (16 lanes x float4 per edge) ----------------
__global__ void gcn_scatter_kernel(const float* __restrict__ H, const int* __restrict__ src,
                                   const int* __restrict__ dst, const float* __restrict__ rdeg_in,
                                   float* __restrict__ acc) {
  int gid = blockIdx.x * blockDim.x + threadIdx.x;
  int e = gid >> 4, lane = gid & 15;
  if (e >= NE) return;
  int s = src[e], d = dst[e];
  float w = rdeg_in[d];
  const float4 hv = ((const float4*)(H + (size_t)s * D))[lane];
  float* ap = acc + (size_t)d * D + 4 * lane;
  atomicAdd(ap + 0, hv.x * w);
  atomicAdd(ap + 1, hv.y * w);
  atomicAdd(ap + 2, hv.z * w);
  atomicAdd(ap + 3, hv.w * w);
}

// ---------------- out = (relu?)(acc + b0 + b1); optionally re-zero acc for next layer ----------------
__global__ void gcn_finish_kernel(float4* __restrict__ acc, const float4* __restrict__ b0,
                                  const float4* __restrict__ b1, float4* __restrict__ out,
                                  int relu_and_rezero) {
  int i = blockIdx.x * blockDim.x + threadIdx.x;          // over NN*D/4
  if (i >= NN * D / 4) return;
  int c = i & (D / 4 - 1);                                 // float4 column within row
  float4 v = acc[i];
  float4 p = b0[c], q = b1[c];
  v.x += p.x + q.x; v.y += p.y + q.y; v.z += p.z + q.z; v.w += p.w + q.w;
  if (relu_and_rezero) {
    v.x = fmaxf(v.x, 0.f); v.y = fmaxf(v.y, 0.f);
    v.z = fmaxf(v.z, 0.f); v.w = fmaxf(v.w, 0.f);
    acc[i] = make_float4(0.f, 0.f, 0.f, 0.f);
  }
  out[i] = v;
}

// ---------------- score[e] = dot(H[u[e]], H[v[e]]) (16 lanes x float4, shfl width-16 reduce) ----------
__global__ void gcn_dot_kernel(const float* __restrict__ H, const int* __restrict__ u,
                               const int* __restrict__ v, float* __restrict__ out) {
  int gid = blockIdx.x * blockDim.x + threadIdx.x;
  int e = gid >> 4, lane = gid & 15;
  if (e >= NE) return;
  const float4 a = ((const float4*)(H + (size_t)u[e] * D))[lane];
  const float4 b = ((const float4*)(H + (size_t)v[e] * D))[lane];
  float p = a.x * b.x + a.y * b.y + a.z * b.z + a.w * b.w;
#pragma unroll
  for (int off = 8; off > 0; off >>= 1) p += __shfl_xor(p, off, 16);
  if (lane == 0) out[e] = p;
}

// ---------------- host side ----------------
extern "C" void kernel_launch(void* const* d_in, const int* in_sizes, int n_in,
                              void* d_out, int out_size, void* d_ws, size_t ws_size,
                              hipStream_t stream) {
  const float* x  = (const float*)d_in[0];
  const int* s0   = (const int*)d_in[1];
  const int* d0   = (const int*)d_in[2];
  const int* s1   = (const int*)d_in[3];
  const int* d1   = (const int*)d_in[4];
  const int* ns   = (const int*)d_in[5];
  const int* nd   = (const int*)d_in[6];
  const float* W10 = (const float*)d_in[7];
  const float* W11 = (const float*)d_in[8];
  const float* W20 = (const float*)d_in[9];
  const float* W21 = (const float*)d_in[10];
  const float* b10 = (const float*)d_in[11];
  const float* b11 = (const float*)d_in[12];
  const float* b20 = (const float*)d_in[13];
  const float* b21 = (const float*)d_in[14];
  float* out = (float*)d_out;

  // workspace layout (floats): rdeg[4*NN] | acc[NN*D] | h[NN*D] | y[NN*D]  (~78.4 MB)
  float* rdeg = (float*)d_ws;
  float* acc  = rdeg + 4 * NN;
  float* h    = acc + (size_t)NN * D;
  float* y    = h + (size_t)NN * D;

  const int T = 256;
  const int zero_n4 = (4 * NN + NN * D) / 4;                // rdeg + acc, contiguous, /4 exact
  const int gemm_blocks = ((NN / 16) * 32 + T - 1) / T;     // one wave per 16-row tile
  const int edge_blocks = (NE + T - 1) / T;
  const int ew_blocks   = (NE * 16 + T - 1) / T;            // 16 lanes per edge
  const int node_blocks = (NN * D / 4 + T - 1) / T;

  // degrees (recomputed every call -> deterministic under graph replay)
  gcn_zero_kernel<<<(zero_n4 + T - 1) / T, T, 0, stream>>>((float4*)rdeg, zero_n4);
  gcn_degree_kernel<<<edge_blocks, T, 0, stream>>>(s0, d0, s1, d1, rdeg);
  gcn_rsqrt_kernel<<<(4 * NN + T - 1) / T, T, 0, stream>>>(rdeg, 4 * NN);

  // layer 1: acc = sum_r scatter_r((x@W1_r)*rdeg_out_r) * rdeg_in_r ; y = relu(acc + b10 + b11)
  gcn_gemm_scale_kernel<<<gemm_blocks, T, 0, stream>>>(x, W10, rdeg + 0 * NN, h);
  gcn_scatter_kernel<<<ew_blocks, T, 0, stream>>>(h, s0, d0, rdeg + 1 * NN, acc);
  gcn_gemm_scale_kernel<<<gemm_blocks, T, 0, stream>>>(x, W11, rdeg + 2 * NN, h);
  gcn_scatter_kernel<<<ew_blocks, T, 0, stream>>>(h, s1, d1, rdeg + 3 * NN, acc);
  gcn_finish_kernel<<<node_blocks, T, 0, stream>>>((float4*)acc, (const float4*)b10,
                                                   (const float4*)b11, (float4*)y, 1);

  // layer 2: same on y, no relu; final features land in h
  gcn_gemm_scale_kernel<<<gemm_blocks, T, 0, stream>>>(y, W20, rdeg + 0 * NN, h);
  gcn_scatter_kernel<<<ew_blocks, T, 0, stream>>>(h, s0, d0, rdeg + 1 * NN, acc);
  gcn_gemm_scale_kernel<<<gemm_blocks, T, 0, stream>>>(y, W21, rdeg + 2 * NN, h);
  gcn_scatter_kernel<<<ew_blocks, T, 0, stream>>>(h, s1, d1, rdeg + 3 * NN, acc);
  gcn_finish_kernel<<<node_blocks, T, 0, stream>>>((float4*)acc, (const float4*)b20,
                                                   (const float4*)b21, (float4*)h, 0);

  // predictor: pos scores then neg scores, concatenated in d_out
  gcn_dot_kernel<<<ew_blocks, T, 0, stream>>>(h, s0, d0, out);
  gcn_dot_kernel<<<ew_blocks, T, 0, stream>>>(h, ns, nd, out + NE);
}